// PAL_84928683311432
// MI455X (gfx1250) — compile-verified
//
#include <hip/hip_runtime.h>
#include <math.h>

// ---------------------------------------------------------------------------
// MI455X / gfx1250 implementation.
//  - SpMM via native f32 global atomics (unsafeAtomicAdd -> global_atomic_add_f32)
//  - wave32 shuffle reductions for norms / dots / softmax
//  - fusion MLP via V_WMMA_F32_16X16X4_F32 (exact f32 matrix math)
// ---------------------------------------------------------------------------

typedef float v2f __attribute__((ext_vector_type(2)));
typedef float v8f __attribute__((ext_vector_type(8)));

#define U_NODES 20000
#define I_NODES 40000
#define B_NODES 10000
#define D_DIM   64
#define M_ITEMS 40
#define BATCH_N 8192
#define K_PER   2

// ---------------- wave32 helpers ----------------
__device__ __forceinline__ float wave_sum32(float v) {
    #pragma unroll
    for (int off = 16; off > 0; off >>= 1) v += __shfl_xor(v, off, 32);
    return v;
}
__device__ __forceinline__ float wave_max32(float v) {
    #pragma unroll
    for (int off = 16; off > 0; off >>= 1) v = fmaxf(v, __shfl_xor(v, off, 32));
    return v;
}

// ---------------- utility kernels ----------------
__global__ __launch_bounds__(256) void k_fill_zero(float* __restrict__ p, int n) {
    int i = blockIdx.x * blockDim.x + threadIdx.x;
    if (i < n) p[i] = 0.0f;
}

// f = acc = concat(A, B) along rows (flat element copy)
__global__ __launch_bounds__(256) void k_concat(const float* __restrict__ A,
                                                const float* __restrict__ B,
                                                float* __restrict__ f,
                                                float* __restrict__ acc,
                                                int nAelem, int total) {
    int i = blockIdx.x * blockDim.x + threadIdx.x;
    if (i < total) {
        float v = (i < nAelem) ? A[i] : B[i - nAelem];
        f[i]   = v;
        acc[i] = v;
    }
}

// COO SpMM with scale folded in: y[row,:] += scale * val * x[col,:]
// 16 threads per edge, 4 dims each (float4 gather, 4 native f32 atomics).
__global__ __launch_bounds__(256) void k_spmm(const int* __restrict__ rows,
                                              const int* __restrict__ cols,
                                              const float* __restrict__ vals,
                                              const float* __restrict__ x,
                                              float* __restrict__ y,
                                              int nnz, float scale) {
    int t = blockIdx.x * blockDim.x + threadIdx.x;
    int e = t >> 4;
    if (e >= nnz) return;
    int c   = (t & 15) * 4;
    int row = rows[e];
    int col = cols[e];
    float v = vals[e] * scale;
    const float4 xv = *(const float4*)(x + (size_t)col * D_DIM + c);
    float* yp = y + (size_t)row * D_DIM + c;
    unsafeAtomicAdd(yp + 0, v * xv.x);
    unsafeAtomicAdd(yp + 1, v * xv.y);
    unsafeAtomicAdd(yp + 2, v * xv.z);
    unsafeAtomicAdd(yp + 3, v * xv.w);
}

// acc[row,:] += f[row,:] / max(||f[row,:]||, 1e-12). One wave32 per row.
__global__ __launch_bounds__(256) void k_norm_acc(const float* __restrict__ f,
                                                  float* __restrict__ acc, int n) {
    int wave = (blockIdx.x * blockDim.x + threadIdx.x) >> 5;
    int lane = threadIdx.x & 31;
    if (wave >= n) return;
    const float* fr = f + (size_t)wave * D_DIM;
    float a = fr[lane];
    float b = fr[lane + 32];
    float ss = wave_sum32(a * a + b * b);
    float inv = 1.0f / fmaxf(sqrtf(ss), 1e-12f);
    float* ar = acc + (size_t)wave * D_DIM;
    ar[lane]      += a * inv;
    ar[lane + 32] += b * inv;
}

// ---------------- attention scoring: one wave per (b,k) pair ----------------
__global__ __launch_bounds__(256) void k_score(const float* __restrict__ accIL,
                                               const float* __restrict__ accBL,
                                               const int* __restrict__ item_ids,
                                               const unsigned char* __restrict__ item_mask,
                                               const int* __restrict__ users,
                                               const int* __restrict__ bundles,
                                               float* __restrict__ scoreIL,
                                               float* __restrict__ scoreBL) {
    const int WPB = 8;                       // 8 wave32 per 256-thread block
    __shared__ float s_w[WPB][M_ITEMS];      // logits -> softmax weights
    __shared__ int   s_id[WPB][M_ITEMS];

    int wslot = threadIdx.x >> 5;
    int lane  = threadIdx.x & 31;
    int pair  = blockIdx.x * WPB + wslot;    // grid sized exactly: no stragglers
    int b      = pair >> 1;
    int u      = users[b];
    int bundle = bundles[pair];

    const float* q = accIL + (size_t)u * D_DIM;
    float q0 = q[lane], q1 = q[lane + 32];

    // stage item ids
    s_id[wslot][lane] = item_ids[(size_t)bundle * M_ITEMS + lane];       // m = 0..31
    if (lane < M_ITEMS - 32)
        s_id[wslot][32 + lane] = item_ids[(size_t)bundle * M_ITEMS + 32 + lane];
    __syncthreads();

    // logits (masked)
    for (int m = 0; m < M_ITEMS; ++m) {
        const float* it = accIL + (size_t)(U_NODES + s_id[wslot][m]) * D_DIM;
        float p = wave_sum32(q0 * it[lane] + q1 * it[lane + 32]);
        if (lane == 0) {
            bool mk = item_mask[(size_t)bundle * M_ITEMS + m] != 0;
            s_w[wslot][m] = mk ? p : -1.0e9f;
        }
    }
    __syncthreads();

    // masked softmax over the 40 logits (lane m, lanes 0..7 also m+32)
    float v0 = s_w[wslot][lane];
    float v1 = (lane < M_ITEMS - 32) ? s_w[wslot][lane + 32] : -1.0e30f;
    float mx = wave_max32(fmaxf(v0, v1));
    float e0 = __expf(v0 - mx);
    float e1 = (lane < M_ITEMS - 32) ? __expf(v1 - mx) : 0.0f;
    float se = wave_sum32(e0 + e1);
    float isum = 1.0f / se;
    __syncthreads();
    s_w[wslot][lane] = e0 * isum;
    if (lane < M_ITEMS - 32) s_w[wslot][lane + 32] = e1 * isum;
    __syncthreads();

    // attended bundle rep + il score
    float ib0 = 0.0f, ib1 = 0.0f;
    for (int m = 0; m < M_ITEMS; ++m) {
        float w = s_w[wslot][m];
        const float* it = accIL + (size_t)(U_NODES + s_id[wslot][m]) * D_DIM;
        ib0 += w * it[lane];
        ib1 += w * it[lane + 32];
    }
    float il_s = wave_sum32(q0 * ib0 + q1 * ib1);

    // bundle-level score
    const float* bu = accBL + (size_t)u * D_DIM;
    const float* bb = accBL + (size_t)(U_NODES + bundle) * D_DIM;
    float bl_s = wave_sum32(bu[lane] * bb[lane] + bu[lane + 32] * bb[lane + 32]);

    if (lane == 0) {
        scoreIL[pair] = il_s;
        scoreBL[pair] = bl_s;
    }
}

// ---------------- fusion gate MLP via V_WMMA_F32_16X16X4_F32 ----------------
// Block = 128 threads (4 wave32), one 16-row tile of the batch.
// Wave w computes H[16, 16w:16w+16] = relu(X[16,128] @ W1[:,16w:16w+16] + b1).
__global__ __launch_bounds__(128) void k_mlp_gate(const float* __restrict__ accIL,
                                                  const float* __restrict__ accBL,
                                                  const int* __restrict__ users,
                                                  const float* __restrict__ fw1,
                                                  const float* __restrict__ fb1,
                                                  const float* __restrict__ fw2,
                                                  const float* __restrict__ fb2,
                                                  const float* __restrict__ scoreIL,
                                                  const float* __restrict__ scoreBL,
                                                  float* __restrict__ out) {
    __shared__ float Xs[16][128];   // gate_in tile
    __shared__ float Hs[16][64];    // hidden tile
    __shared__ int   su[16];

    const int tid = threadIdx.x;
    const int b0  = blockIdx.x * 16;

    if (tid < 16) su[tid] = users[b0 + tid];
    __syncthreads();

    // gather gate_in = [IL_u[users] | BL_u[users]]
    for (int idx = tid; idx < 16 * 128; idx += 128) {
        int r = idx >> 7, c = idx & 127;
        int u = su[r];
        Xs[r][c] = (c < 64) ? accIL[(size_t)u * D_DIM + c]
                            : accBL[(size_t)u * D_DIM + (c - 64)];
    }
    __syncthreads();

    const int wave = tid >> 5;
    const int lane = tid & 31;
    const int l15  = lane & 15;
    const int hi   = lane >> 4;      // 0 for lanes 0-15, 1 for 16-31
    const int n0   = wave * 16;

    v8f c = {};
    #pragma unroll
    for (int k0 = 0; k0 < 128; k0 += 4) {
        const int ka = k0 + 2 * hi;  // A: lanes 0-15 -> K=k0,k0+1 ; 16-31 -> K=k0+2,k0+3
        v2f a, bm;
        a.x  = Xs[l15][ka];
        a.y  = Xs[l15][ka + 1];
        bm.x = fw1[(size_t)ka * 64 + n0 + l15];        // B: same K split, N across lanes
        bm.y = fw1[(size_t)(ka + 1) * 64 + n0 + l15];
        c = __builtin_amdgcn_wmma_f32_16x16x4_f32(
                /*neg_a=*/false, a, /*neg_b=*/false, bm,
                /*c_mod=*/(short)0, c, /*reuse_a=*/false, /*reuse_b=*/false);
    }

    // bias + relu; D layout: VGPR j -> M = j (lanes 0-15) / j+8 (lanes 16-31)
    const float bias = fb1[n0 + l15];
    #pragma unroll
    for (int j = 0; j < 8; ++j) {
        int m = hi ? (j + 8) : j;
        float v = c[j] + bias;
        Hs[m][n0 + l15] = v > 0.0f ? v : 0.0f;
    }
    __syncthreads();

    // H @ fw2 + fb2 -> sigmoid -> blend scores
    if (tid < 16) {
        float s = 0.0f;
        #pragma unroll
        for (int n = 0; n < 64; ++n) s += Hs[tid][n] * fw2[n];
        s += fb2[0];
        float g = 1.0f / (1.0f + __expf(-s));
        int b = b0 + tid;
        out[b * K_PER + 0] = g * scoreIL[b * K_PER + 0] + (1.0f - g) * scoreBL[b * K_PER + 0];
        out[b * K_PER + 1] = g * scoreIL[b * K_PER + 1] + (1.0f - g) * scoreBL[b * K_PER + 1];
    }
}

// ---------------------------------------------------------------------------
extern "C" void kernel_launch(void* const* d_in, const int* in_sizes, int n_in,
                              void* d_out, int out_size, void* d_ws, size_t ws_size,
                              hipStream_t stream) {
    const float* users_feature   = (const float*)d_in[0];
    const float* items_feature   = (const float*)d_in[1];
    const float* bundles_feature = (const float*)d_in[2];
    const float* fw1             = (const float*)d_in[3];
    const float* fb1             = (const float*)d_in[4];
    const float* fw2             = (const float*)d_in[5];
    const float* fb2             = (const float*)d_in[6];
    const int*   il_rows         = (const int*)d_in[7];
    const int*   il_cols         = (const int*)d_in[8];
    const float* il_vals         = (const float*)d_in[9];
    const int*   bl_rows         = (const int*)d_in[10];
    const int*   bl_cols         = (const int*)d_in[11];
    const float* bl_vals         = (const float*)d_in[12];
    const int*   bundle_item_ids = (const int*)d_in[13];
    const unsigned char* bundle_item_mask = (const unsigned char*)d_in[14];
    const int*   users           = (const int*)d_in[15];
    const int*   bundles         = (const int*)d_in[16];
    float*       out             = (float*)d_out;

    const int nnzIL = in_sizes[7];
    const int nnzBL = in_sizes[10];

    // workspace layout (floats)
    const size_t NIL = (size_t)(U_NODES + I_NODES) * D_DIM;  // 3,840,000
    const size_t NBL = (size_t)(U_NODES + B_NODES) * D_DIM;  // 1,920,000
    float* ws      = (float*)d_ws;
    float* accIL   = ws;
    float* accBL   = accIL + NIL;
    float* fcur    = accBL + NBL;      // sized for larger graph
    float* fnext   = fcur + NIL;
    float* scoreIL = fnext + NIL;
    float* scoreBL = scoreIL + (size_t)BATCH_N * K_PER;

    auto cdiv = [](long long a, long long b) { return (unsigned)((a + b - 1) / b); };

    // ---- one graph propagation (L=2 layers, acc = f0 + f1/|f1| + f2/|f2|) ----
    auto run_graph = [&](const int* rows, const int* cols, const float* vals, int nnz,
                         const float* featB, int nB, float* acc) {
        const int n     = U_NODES + nB;
        const int total = n * D_DIM;
        k_concat<<<cdiv(total, 256), 256, 0, stream>>>(
            users_feature, featB, fcur, acc, U_NODES * D_DIM, total);
        k_fill_zero<<<cdiv(total, 256), 256, 0, stream>>>(fnext, total);
        k_spmm<<<cdiv((long long)nnz * 16, 256), 256, 0, stream>>>(
            rows, cols, vals, fcur, fnext, nnz, 1.0f / 2.0f);
        k_norm_acc<<<cdiv(n, 8), 256, 0, stream>>>(fnext, acc, n);
        k_fill_zero<<<cdiv(total, 256), 256, 0, stream>>>(fcur, total);
        k_spmm<<<cdiv((long long)nnz * 16, 256), 256, 0, stream>>>(
            rows, cols, vals, fnext, fcur, nnz, 1.0f / 3.0f);
        k_norm_acc<<<cdiv(n, 8), 256, 0, stream>>>(fcur, acc, n);
    };

    run_graph(il_rows, il_cols, il_vals, nnzIL, items_feature,   I_NODES, accIL);
    run_graph(bl_rows, bl_cols, bl_vals, nnzBL, bundles_feature, B_NODES, accBL);

    // ---- attention scoring: 16384 pairs, 8 waves per block -> 2048 blocks ----
    k_score<<<(BATCH_N * K_PER) / 8, 256, 0, stream>>>(
        accIL, accBL, bundle_item_ids, bundle_item_mask, users, bundles,
        scoreIL, scoreBL);

    // ---- fusion gate MLP (WMMA) + final blend: 512 blocks x 128 threads ----
    k_mlp_gate<<<BATCH_N / 16, 128, 0, stream>>>(
        accIL, accBL, users, fw1, fb1, fw2, fb2, scoreIL, scoreBL, out);
}